// GraphMomentAggregator_33139967655986
// MI455X (gfx1250) — compile-verified
//
#include <hip/hip_runtime.h>
#include <hip/hip_bf16.h>

#define NUM_GRAPHS 1024
#define F_DIM 64
#define S_DIM 4
#define NCOLS (F_DIM * S_DIM)   // 256 output columns per graph
#define LSTR  260               // LDS row stride (bank-skewing pad)

typedef float v2f __attribute__((ext_vector_type(2)));
typedef float v8f __attribute__((ext_vector_type(8)));

#if defined(__gfx1250__) && __has_builtin(__builtin_amdgcn_wmma_f32_16x16x4_f32)
#define HAVE_WMMA_F32X4 1
#else
#define HAVE_WMMA_F32X4 0
#endif

// ---------------------------------------------------------------------------
// Kernel 1: segment bounds via binary search (sorted batch_index, int64).
// seg[g] = first row with batch >= g, for g in [0,1024]; seg[1024] = N.
// ---------------------------------------------------------------------------
__global__ void seg_bounds_kernel(const long long* __restrict__ bidx,
                                  int* __restrict__ seg, int N) {
    int g = blockIdx.x * blockDim.x + threadIdx.x;
    if (g > NUM_GRAPHS) return;
    int lo = 0, hi = N;
    while (lo < hi) {
        int mid = (lo + hi) >> 1;
        if (bidx[mid] < (long long)g) lo = mid + 1; else hi = mid;
    }
    seg[g] = lo;
}

// ---------------------------------------------------------------------------
// Hot streaming loop: per-thread accumulation of 4 features x 4 moments.
// DO_ABS is a uniform compile-time branch (no per-element selects).
// fabsf folds into VALU input modifiers; p4 = p3*b matches reference cumprod.
// ---------------------------------------------------------------------------
template <bool DO_ABS>
__device__ __forceinline__ void row_loop(const float* __restrict__ x,
                                         int s0, int s1, int rlane, int f4,
                                         float (&acc)[4][4]) {
    for (int r = s0 + rlane; r < s1; r += 16) {
        const float4 v = *reinterpret_cast<const float4*>(
            x + (size_t)r * F_DIM + f4 * 4);
        // stream-ahead prefetch (-> global_prefetch_b8 on gfx1250)
        __builtin_prefetch(x + (size_t)(r + 64) * F_DIM + f4 * 4, 0, 1);
        const float e[4] = {v.x, v.y, v.z, v.w};
#pragma unroll
        for (int j = 0; j < 4; ++j) {
            const float b  = DO_ABS ? __builtin_fabsf(e[j]) : e[j];
            const float p2 = b * b;
            const float p3 = p2 * b;
            const float p4 = p3 * b;
            acc[j][0] += b;  acc[j][1] += p2;
            acc[j][2] += p3; acc[j][3] += p4;
        }
    }
}

// ---------------------------------------------------------------------------
// Kernel 2: one block (8 wave32 waves) per graph. Coalesced 128-bit loads,
// register accumulation, then a 16-way cross-row-lane reduction done with
// V_WMMA_F32_16X16X4_F32 (A = all-ones => exact fp32 column sums).
// ---------------------------------------------------------------------------
__global__ __launch_bounds__(256)
void moments_kernel(const float* __restrict__ x,
                    const int* __restrict__ seg,
                    const int* __restrict__ abs_flag,
                    float* __restrict__ out, int N) {
    const int g     = blockIdx.x;
    const int tid   = threadIdx.x;
    const int rlane = tid >> 4;   // 16 row lanes
    const int f4    = tid & 15;   // float4 feature group

    const int s0 = seg[g];
    const int s1 = seg[g + 1];

    float acc[4][4] = {};
    if (abs_flag[0] != 0) row_loop<true >(x, s0, s1, rlane, f4, acc);
    else                  row_loop<false>(x, s0, s1, rlane, f4, acc);

    // Stage partials: lds_p[rlane][col], col = f*4 + s = f4*16 + j*4 + s.
    __shared__ float lds_p[16][LSTR];
#pragma unroll
    for (int j = 0; j < 4; ++j)
#pragma unroll
        for (int s = 0; s < S_DIM; ++s)
            lds_p[rlane][f4 * 16 + j * 4 + s] = acc[j][s];
    __syncthreads();

#if HAVE_WMMA_F32X4
    // 16-way reduction over row lanes: D = ones(16x4) * B(4x16) + C.
    // All rows of D equal the column sums; A==1 makes the result invariant
    // to the K->(lane-half,VGPR) mapping. EXEC is all-ones here (no
    // divergence before this point in the block).
    const int lane  = tid & 31;
    const int wv    = tid >> 5;       // 8 waves, 32 cols each
    const int halfk = lane >> 4;
    const int nidx  = lane & 15;
    const v2f ones  = {1.0f, 1.0f};
#pragma unroll
    for (int t = 0; t < 2; ++t) {
        const int cbase = wv * 32 + t * 16;
        v8f c8 = {0.f, 0.f, 0.f, 0.f, 0.f, 0.f, 0.f, 0.f};
#pragma unroll
        for (int kb = 0; kb < 16; kb += 4) {
            v2f b;
            b.x = lds_p[kb + halfk][cbase + nidx];
            b.y = lds_p[kb + 2 + halfk][cbase + nidx];
            c8 = __builtin_amdgcn_wmma_f32_16x16x4_f32(
                false, ones, false, b, (short)0, c8, false, false);
        }
        if (lane < 16)  // VGPR0, lanes 0..15 = row M=0, N = lane
            out[(size_t)g * NCOLS + cbase + lane] = c8[0];
    }
#else
    // Fallback: scalar LDS tree (also used for the host compilation pass).
    const int col = tid;
    float sum = 0.f;
#pragma unroll
    for (int k = 0; k < 16; ++k) sum += lds_p[k][col];
    out[(size_t)g * NCOLS + col] = sum;
#endif
}

// ---------------------------------------------------------------------------
extern "C" void kernel_launch(void* const* d_in, const int* in_sizes, int n_in,
                              void* d_out, int out_size, void* d_ws, size_t ws_size,
                              hipStream_t stream) {
    const float*     x    = (const float*)d_in[0];
    const long long* bidx = (const long long*)d_in[1];
    const int*       aflg = (const int*)d_in[2];
    float*           out  = (float*)d_out;
    const int        N    = in_sizes[1];

    int* seg = (int*)d_ws;  // 1025 ints of scratch, rewritten every call

    seg_bounds_kernel<<<(NUM_GRAPHS + 1 + 255) / 256, 256, 0, stream>>>(bidx, seg, N);
    moments_kernel<<<NUM_GRAPHS, 256, 0, stream>>>(x, seg, aflg, out, N);
}